// MeshSDFLoss_89412629168593
// MI455X (gfx1250) — compile-verified
//
#include <hip/hip_runtime.h>
#include <hip/hip_bf16.h>

typedef __attribute__((ext_vector_type(2))) float v2f;
typedef __attribute__((ext_vector_type(8))) float v8f;

#define EPSF 1e-12f

__device__ __forceinline__ float safef(float x) {
    return (fabsf(x) > EPSF) ? x : EPSF;
}

__device__ __forceinline__ v8f wmma_k4(v2f a, v2f b) {
    v8f c = {0.f, 0.f, 0.f, 0.f, 0.f, 0.f, 0.f, 0.f};
    // D = A(16x4,f32) x B(4x16,f32) + 0  -> v_wmma_f32_16x16x4_f32
    return __builtin_amdgcn_wmma_f32_16x16x4_f32(false, a, false, b,
                                                 (short)0, c, false, false);
}

// ---------------- Workspace layout (floats) ---------------------------------
//  [0,4F)    : Bab rows K0..K3  (abx | aby | abz | zeros), stride F per row
//  [4F,8F)   : Bac rows K0..K3  (acx | acy | acz | zeros)
//  [8F,12F)  : Ba  rows K0..K3  (ax  | ay  | az  | zeros)
//  [12F,24F) : per-face consts, 12 floats AoS (48B, b128-friendly):
//              {abab, acac, abac, aab, aac, aa, r_abab, r_acac, r_bc, 0,0,0}
//  [24F,24F+N): pp = |p|^2
__global__ void mesh_prep(const float* __restrict__ verts,
                          const long long* __restrict__ faces,
                          const float* __restrict__ pts,
                          float* __restrict__ ws, int F, int N) {
    int i = blockIdx.x * blockDim.x + threadIdx.x;
    if (i < F) {
        long long i0 = faces[3 * i + 0];
        long long i1 = faces[3 * i + 1];
        long long i2 = faces[3 * i + 2];
        float ax = verts[3 * i0 + 0], ay = verts[3 * i0 + 1], az = verts[3 * i0 + 2];
        float bx = verts[3 * i1 + 0], by = verts[3 * i1 + 1], bz = verts[3 * i1 + 2];
        float cx = verts[3 * i2 + 0], cy = verts[3 * i2 + 1], cz = verts[3 * i2 + 2];
        float abx = bx - ax, aby = by - ay, abz = bz - az;
        float acx = cx - ax, acy = cy - ay, acz = cz - az;

        // B operand blocks (K3 row = zeros -> no cndmask in hot loop)
        ws[0 * F + i] = abx;  ws[1 * F + i] = aby;  ws[2 * F + i] = abz;  ws[3 * F + i] = 0.f;
        ws[4 * F + i] = acx;  ws[5 * F + i] = acy;  ws[6 * F + i] = acz;  ws[7 * F + i] = 0.f;
        ws[8 * F + i] = ax;   ws[9 * F + i] = ay;   ws[10 * F + i] = az;  ws[11 * F + i] = 0.f;

        float abab = abx * abx + aby * aby + abz * abz;
        float acac = acx * acx + acy * acy + acz * acz;
        float abac = abx * acx + aby * acy + abz * acz;
        float aab  = ax * abx + ay * aby + az * abz;
        float aac  = ax * acx + ay * acy + az * acz;
        float aa   = ax * ax + ay * ay + az * az;
        float bc   = abab - 2.f * abac + acac;       // |b-c|^2

        float* cst = ws + 12 * F + 12 * i;
        cst[0] = abab;  cst[1] = acac;  cst[2] = abac;
        cst[3] = aab;   cst[4] = aac;   cst[5] = aa;
        cst[6] = 1.f / safef(abab);     // t_ab denominator (== d1-d3)
        cst[7] = 1.f / safef(acac);     // t_ac denominator (== d2-d6)
        cst[8] = 1.f / safef(bc);       // t_bc denominator
        cst[9] = 0.f; cst[10] = 0.f; cst[11] = 0.f;
    }
    if (i < N) {
        float px = pts[3 * i + 0], py = pts[3 * i + 1], pz = pts[3 * i + 2];
        ws[24 * F + i] = px * px + py * py + pz * pz;
    }
}

// ---------------- Main: WMMA dot products + region clamp + min/argmin --------
__global__ void mesh_main(const float* __restrict__ pts,
                          const float* __restrict__ ws,
                          float* __restrict__ out, int F, int N) {
    const int lane = threadIdx.x & 31;
    const int wid  = (blockIdx.x * blockDim.x + threadIdx.x) >> 5;
    const int pbase = wid * 16;
    if (pbase >= N) return;              // wave-uniform; EXEC stays all-ones

    const int m  = lane & 15;            // A-matrix row / B-matrix column
    const int hi = lane >> 4;            // K half selector

    // A matrix (points): VGPR0 = K0 (x) / K2 (z); VGPR1 = K1 (y) / K3 (0)
    int pm = pbase + m; if (pm > N - 1) pm = N - 1;
    const float px = pts[3 * pm + 0];
    const float py = pts[3 * pm + 1];
    const float pz = pts[3 * pm + 2];
    v2f Apt;
    Apt.x = hi ? pz : px;
    Apt.y = hi ? 0.f : py;

    float ppv[8];
#pragma unroll
    for (int r = 0; r < 8; ++r) {
        int pr = pbase + r + hi * 8; if (pr > N - 1) pr = N - 1;
        ppv[r] = ws[24 * F + pr];
    }

    float best[8];
    int   bidx[8];
#pragma unroll
    for (int r = 0; r < 8; ++r) { best[r] = 3.4e38f; bidx[r] = 0; }

    // per-lane B row bases: row (2*hi) for .x, row (2*hi+1) for .y
    const float* bab0 = ws + (2 * hi + 0) * F;
    const float* bab1 = ws + (2 * hi + 1) * F;
    const float* bac0 = bab0 + 4 * F;
    const float* bac1 = bab1 + 4 * F;
    const float* ba0  = bab0 + 8 * F;
    const float* ba1  = bab1 + 8 * F;
    const float4* cst = (const float4*)(ws + 12 * F);

    const int ntiles = F >> 4;
    for (int ft = 0; ft < ntiles; ++ft) {
        const int f = (ft << 4) + m;     // face this lane represents (col n)

        v2f Bab; Bab.x = bab0[f]; Bab.y = bab1[f];
        v2f Bac; Bac.x = bac0[f]; Bac.y = bac1[f];
        v2f Ba;  Ba.x  = ba0[f];  Ba.y  = ba1[f];

        float4 c0 = cst[3 * f + 0];      // abab, acac, abac, aab
        float4 c1 = cst[3 * f + 1];      // aac, aa, r_ab, r_ac
        float4 c2 = cst[3 * f + 2];      // r_bc, 0, 0, 0
        const float abab = c0.x, acac = c0.y, abac = c0.z, aab = c0.w;
        const float aac = c1.x, aa = c1.y, r_ab = c1.z, r_ac = c1.w;
        const float r_bc = c2.x;

        v8f Cab = wmma_k4(Apt, Bab);     // p . ab
        v8f Cac = wmma_k4(Apt, Bac);     // p . ac
        v8f Cpa = wmma_k4(Apt, Ba);      // p . a

#pragma unroll
        for (int r = 0; r < 8; ++r) {
            float d1 = Cab[r] - aab;
            float d2 = Cac[r] - aac;
            float d3 = d1 - abab;
            float d4 = d2 - abac;
            float d5 = d1 - abac;
            float d6 = d2 - acac;
            float apap = fmaf(-2.f, Cpa[r], ppv[r]) + aa;

            float va = d3 * d6 - d5 * d4;
            float vb = d5 * d2 - d1 * d6;
            float vc = d1 * d4 - d3 * d2;

            float t_ab = d1 * r_ab;
            float t_ac = d2 * r_ac;
            float t_bc = (d4 - d3) * r_bc;

            float rs = __builtin_amdgcn_rcpf(safef(va + vb + vc));
            float v = vb * rs;
            float w = vc * rs;

            bool cBC = (va <= 0.f) && (d4 - d3 >= 0.f) && (d5 - d6 >= 0.f);
            v = cBC ? 1.f - t_bc : v;  w = cBC ? t_bc : w;
            bool cAC = (vb <= 0.f) && (d2 >= 0.f) && (d6 <= 0.f);
            v = cAC ? 0.f : v;         w = cAC ? t_ac : w;
            bool cC = (d6 >= 0.f) && (d5 <= d6);
            v = cC ? 0.f : v;          w = cC ? 1.f : w;
            bool cAB = (vc <= 0.f) && (d1 >= 0.f) && (d3 <= 0.f);
            v = cAB ? t_ab : v;        w = cAB ? 0.f : w;
            bool cB = (d3 >= 0.f) && (d4 <= d3);
            v = cB ? 1.f : v;          w = cB ? 0.f : w;
            bool cA = (d1 <= 0.f) && (d2 <= 0.f);
            v = cA ? 0.f : v;          w = cA ? 0.f : w;

            float sq = apap - 2.f * v * d1 - 2.f * w * d2
                     + v * v * abab + 2.f * v * w * abac + w * w * acac;
            sq = fmaxf(sq, 0.f);

            if (sq < best[r]) { best[r] = sq; bidx[r] = f; }
        }
    }

    // faces live across lanes within each 16-lane half: butterfly min+argmin
#pragma unroll
    for (int r = 0; r < 8; ++r) {
        float s = best[r];
        int   bi = bidx[r];
#pragma unroll
        for (int off = 1; off < 16; off <<= 1) {
            float os = __shfl_xor(s, off, 16);
            int   oi = __shfl_xor(bi, off, 16);
            if (os < s || (os == s && oi < bi)) { s = os; bi = oi; }
        }
        if ((lane & 15) == 0) {
            int M = r + hi * 8;
            if (pbase + M < N) {
                out[1 + pbase + M]     = s;               // dist
                out[1 + N + pbase + M] = (float)bi;       // assoc
            }
        }
    }
}

// ---------------- Deterministic loss reduction -------------------------------
__global__ void mesh_loss(const float* __restrict__ dist, float* __restrict__ out, int N) {
    __shared__ float sm[256];
    float s = 0.f;
    for (int i = threadIdx.x; i < N; i += 256) s += dist[i];
    sm[threadIdx.x] = s;
    __syncthreads();
    for (int st = 128; st > 0; st >>= 1) {
        if ((int)threadIdx.x < st) sm[threadIdx.x] += sm[threadIdx.x + st];
        __syncthreads();
    }
    if (threadIdx.x == 0) out[0] = sm[0] / (float)N;
}

extern "C" void kernel_launch(void* const* d_in, const int* in_sizes, int n_in,
                              void* d_out, int out_size, void* d_ws, size_t ws_size,
                              hipStream_t stream) {
    const float*     verts = (const float*)d_in[0];
    const long long* faces = (const long long*)d_in[1];
    const float*     pts   = (const float*)d_in[2];
    float* out = (float*)d_out;
    float* ws  = (float*)d_ws;

    const int F = in_sizes[1] / 3;   // 8192
    const int N = in_sizes[2] / 3;   // 2048

    int prepN = (F > N ? F : N);
    mesh_prep<<<(prepN + 255) / 256, 256, 0, stream>>>(verts, faces, pts, ws, F, N);

    int waves   = (N + 15) / 16;     // 128 waves
    int threads = waves * 32;
    int block   = 128;               // 4 waves / workgroup
    mesh_main<<<(threads + block - 1) / block, block, 0, stream>>>(pts, ws, out, F, N);

    mesh_loss<<<1, 256, 0, stream>>>(out + 1, out, N);
}